// GCN_74148315398313
// MI455X (gfx1250) — compile-verified
//
#include <hip/hip_runtime.h>
#include <hip/hip_bf16.h>

typedef float v2f __attribute__((ext_vector_type(2)));
typedef float v8f __attribute__((ext_vector_type(8)));

#define IN_D  64
#define HID_D 64
#define OUT_D 32

// ---------------------------------------------------------------------------
// Degree / normalization kernels
// ---------------------------------------------------------------------------
__global__ void deg_init(float* __restrict__ deg, int n) {
    int i = blockIdx.x * blockDim.x + threadIdx.x;
    if (i < n) deg[i] = 1.0f;              // self-loop contributes 1
}

__global__ void deg_count(const int* __restrict__ dst, float* __restrict__ deg, int e) {
    int i = blockIdx.x * blockDim.x + threadIdx.x;
    if (i < e) atomicAdd(&deg[dst[i]], 1.0f);
}

__global__ void deg_rsqrt(float* __restrict__ deg, int n) {
    int i = blockIdx.x * blockDim.x + threadIdx.x;
    if (i < n) deg[i] = rsqrtf(deg[i]);    // deg >= 1 always (self-loop)
}

// ---------------------------------------------------------------------------
// fp32 WMMA GEMM:  Y[nrows x DOUT] = X[nrows x 64] * W[64 x DOUT]
// One wave per 16-row tile; 8 waves per block; W staged in LDS.
//
// A-matrix 16x4 fp32 layout (ISA 7.12.2): lanes 0-15 hold K=0,1 in v[0],v[1];
// lanes 16-31 hold K=2,3. B 4x16 mirrored; C/D 16x16: VGPR r -> row r (lanes
// 0-15) and row r+8 (lanes 16-31), col = lane&15.
//
// W is staged into LDS pair-interleaved:
//   sW[((k>>1)*DOUT + col)*2 + (k&1)] = W[k*DOUT + col]
// so each lane's B operand (W[k][col], W[k+1][col]) is one aligned 8-byte
// word -> a single ds_load_b64 into a contiguous VGPR pair (no mov shuffles).
// ---------------------------------------------------------------------------
template<int DOUT>
__global__ __launch_bounds__(256) void gemm_wmma_f32(
        const float* __restrict__ X, const float* __restrict__ W,
        float* __restrict__ Y, int nrows) {
    __shared__ float sW[64 * DOUT];
    for (int i = threadIdx.x; i < 64 * DOUT; i += 256) {
        int k   = i / DOUT;
        int col = i - k * DOUT;
        sW[((k >> 1) * DOUT + col) * 2 + (k & 1)] = W[i];
    }
    __syncthreads();

    const int lane = threadIdx.x & 31;
    const int wave = threadIdx.x >> 5;
    const int m    = lane & 15;
    const int khi  = lane >> 4;            // 0 for lanes 0-15, 1 for 16-31
    const int rowBase = (blockIdx.x * 8 + wave) * 16;
    if (rowBase >= nrows) return;          // wave-uniform: EXEC stays all-1s

    v8f acc[DOUT / 16];
#pragma unroll
    for (int t = 0; t < DOUT / 16; ++t) acc[t] = (v8f)0.0f;

    const float* xrow = X + (size_t)(rowBase + m) * 64;
    const v2f*   sWp  = (const v2f*)sW;    // pair-interleaved view
#pragma unroll
    for (int k4 = 0; k4 < 64; k4 += 4) {
        // A pair: contiguous in memory -> global_load_b64
        v2f a = *(const v2f*)(xrow + k4 + 2 * khi);
        const int kp = (k4 >> 1) + khi;    // pair row index for this lane
#pragma unroll
        for (int t = 0; t < DOUT / 16; ++t) {
            // B pair: one aligned ds_load_b64
            v2f b = sWp[kp * DOUT + t * 16 + m];
            acc[t] = __builtin_amdgcn_wmma_f32_16x16x4_f32(
                false, a, false, b, (short)0, acc[t], false, false);
        }
    }

#pragma unroll
    for (int t = 0; t < DOUT / 16; ++t)
#pragma unroll
        for (int r = 0; r < 8; ++r)
            Y[(size_t)(rowBase + r + 8 * khi) * DOUT + t * 16 + m] = acc[t][r];
}

// ---------------------------------------------------------------------------
// Aggregation kernels
// ---------------------------------------------------------------------------
// agg[n*D + f] = bias[f]   (bias folded into the accumulator init)
template<int D>
__global__ void agg_init_bias(float* __restrict__ agg, const float* __restrict__ b, int n) {
    size_t i = (size_t)blockIdx.x * blockDim.x + threadIdx.x;
    if (i < (size_t)n * D) agg[i] = b[i & (D - 1)];
}

// Edge-parallel scatter-add: one (edge, feature) pair per thread. Feature id
// lives in the low bits, so each wave works on a single edge -> edge index /
// dinv loads are wave-uniform (scalar-broadcast), h/agg accesses coalesced.
template<int D>
__global__ void edge_scatter(const int* __restrict__ src, const int* __restrict__ dst,
                             const float* __restrict__ dinv,
                             const float* __restrict__ h, float* __restrict__ agg,
                             int e) {
    size_t i = (size_t)blockIdx.x * blockDim.x + threadIdx.x;
    if (i >= (size_t)e * D) return;
    int ed = (int)(i / D);
    int f  = (int)(i & (D - 1));
    int s  = src[ed];
    int d  = dst[ed];
    float w = dinv[s] * dinv[d];
    atomicAdd(&agg[(size_t)d * D + f], h[(size_t)s * D + f] * w);
}

// Self-loop contribution (+ optional ReLU). Runs after the edge kernel, one
// thread per output element -> no atomics needed.
template<int D, bool RELU>
__global__ void self_loop(const float* __restrict__ h, float* __restrict__ agg,
                          const float* __restrict__ dinv, int n) {
    size_t i = (size_t)blockIdx.x * blockDim.x + threadIdx.x;
    if (i >= (size_t)n * D) return;
    int nd = (int)(i / D);
    float w = dinv[nd] * dinv[nd];
    float v = agg[i] + h[i] * w;
    agg[i] = RELU ? (v > 0.0f ? v : 0.0f) : v;
}

// ---------------------------------------------------------------------------
extern "C" void kernel_launch(void* const* d_in, const int* in_sizes, int n_in,
                              void* d_out, int out_size, void* d_ws, size_t ws_size,
                              hipStream_t stream) {
    const float* x  = (const float*)d_in[0];
    const int*   ei = (const int*)d_in[1];
    const float* W1 = (const float*)d_in[2];
    const float* b1 = (const float*)d_in[3];
    const float* W2 = (const float*)d_in[4];
    const float* b2 = (const float*)d_in[5];
    float* out = (float*)d_out;

    const int N = in_sizes[0] / IN_D;
    const int E = in_sizes[1] / 2;
    const int* src = ei;          // edge_index[0]
    const int* dst = ei + E;      // edge_index[1]

    // Workspace layout (floats): dinv[N] | h1[N*64] | agg[N*64]; h2 reuses h1.
    float* dinv = (float*)d_ws;
    float* h1   = dinv + N;
    float* agg  = h1 + (size_t)N * HID_D;
    float* h2   = h1;             // h1 dead after layer-1 aggregation

    const int T = 256;
    auto g = [](long long n, int t) { return (unsigned)((n + t - 1) / t); };

    // --- normalization -----------------------------------------------------
    deg_init <<<g(N, T), T, 0, stream>>>(dinv, N);
    deg_count<<<g(E, T), T, 0, stream>>>(dst, dinv, E);
    deg_rsqrt<<<g(N, T), T, 0, stream>>>(dinv, N);

    const unsigned gemmBlocks = g((long long)(N + 15) / 16, 8);

    // --- layer 1: h1 = x @ W1 ; agg = relu(scatter(h1) + b1) ---------------
    gemm_wmma_f32<HID_D><<<gemmBlocks, T, 0, stream>>>(x, W1, h1, N);
    agg_init_bias<HID_D><<<g((long long)N * HID_D, T), T, 0, stream>>>(agg, b1, N);
    edge_scatter<HID_D><<<g((long long)E * HID_D, T), T, 0, stream>>>(src, dst, dinv, h1, agg, E);
    self_loop<HID_D, true><<<g((long long)N * HID_D, T), T, 0, stream>>>(h1, agg, dinv, N);

    // --- layer 2: h2 = agg @ W2 ; out = scatter(h2) + b2 -------------------
    gemm_wmma_f32<OUT_D><<<gemmBlocks, T, 0, stream>>>(agg, W2, h2, N);
    agg_init_bias<OUT_D><<<g((long long)N * OUT_D, T), T, 0, stream>>>(out, b2, N);
    edge_scatter<OUT_D><<<g((long long)E * OUT_D, T), T, 0, stream>>>(src, dst, dinv, h2, out, E);
    self_loop<OUT_D, false><<<g((long long)N * OUT_D, T), T, 0, stream>>>(h2, out, dinv, N);
}